// CTCLossFromScratch_77575699300982
// MI455X (gfx1250) — compile-verified
//
#include <hip/hip_runtime.h>
#include <hip/hip_bf16.h>

// ---------------------------------------------------------------------------
// CTC loss forward on MI455X (gfx1250).
// Streaming-bound problem: one workgroup per batch row, double-buffered
// global->LDS async copies (ASYNCcnt) feed a wave32 DP recurrence over S=129
// extended states. Final mean via V_WMMA_F32_16X16X4_F32 ones-reduction.
// ---------------------------------------------------------------------------

#define TPB     256          // 8 wave32 per workgroup
#define MAX_S   129          // 2*Lmax+1 with Lmax=64
#define MAX_C   1024         // class count (reference-fixed)
#define BLANK_  0
#define NEG_INF_F (-1.0e30f)

typedef __attribute__((ext_vector_type(2))) float v2f;
typedef __attribute__((ext_vector_type(8))) float v8f;
typedef int vi4 __attribute__((__vector_size__(4 * sizeof(int))));

// ---- CDNA5 async global->LDS (ASYNCcnt) -----------------------------------
__device__ __forceinline__ void async_load_b128(const float* g, float* l) {
#if __has_builtin(__builtin_amdgcn_global_load_async_to_lds_b128)
  __builtin_amdgcn_global_load_async_to_lds_b128(
      (__attribute__((address_space(1))) vi4*)(unsigned long long)g,
      (__attribute__((address_space(3))) vi4*)(unsigned)(unsigned long long)l,
      0, 0);
#else
  unsigned lds_addr = (unsigned)(unsigned long long)l;
  asm volatile("global_load_async_to_lds_b128 %0, %1, off"
               :: "v"(lds_addr), "v"(g) : "memory");
#endif
}

__device__ __forceinline__ void wait_async_le(int n) {
#if __has_builtin(__builtin_amdgcn_s_wait_asynccnt)
  if (n == 0) __builtin_amdgcn_s_wait_asynccnt(0);
  else        __builtin_amdgcn_s_wait_asynccnt(1);
#else
  if (n == 0) asm volatile("s_wait_asynccnt 0x0" ::: "memory");
  else        asm volatile("s_wait_asynccnt 0x1" ::: "memory");
#endif
}

// jnp.logaddexp: max + log1p(exp(min-max)); exact for the -1e30 sentinel in fp32
__device__ __forceinline__ float lae(float a, float b) {
  float m = fmaxf(a, b);
  float d = fminf(a, b) - m;
  return m + log1pf(expf(d));
}

// ---------------------------------------------------------------------------
// Per-batch alpha recursion. Grid = B blocks of TPB threads.
// ---------------------------------------------------------------------------
__global__ __launch_bounds__(TPB)
void CTC_alpha_kernel(const float* __restrict__ log_probs,   // [T,B,C]
                      const int*   __restrict__ targets,     // [B*Lmax]
                      const int*   __restrict__ input_lengths,
                      const int*   __restrict__ target_lengths,
                      float*       __restrict__ nll_out,     // [B]
                      int T, int B, int C, int Lmax) {
  __shared__ alignas(16) float rowbuf[2][MAX_C];   // double-buffered emission rows
  __shared__ float alpha[MAX_S];
  __shared__ int   ext[MAX_S];
  __shared__ int   skipc[MAX_S];
  __shared__ int   offset_sh;

  const int b   = blockIdx.x;
  const int tid = threadIdx.x;
  const int S   = 2 * Lmax + 1;
  const int Tb  = input_lengths[b];     // freeze at t >= Tb == stop early
  const int Lb  = target_lengths[b];

  // --- kick off async stream of rows 0 and 1 (one b128 instr per wave each)
  const float* row0 = log_probs + ((size_t)0 * B + b) * C;
  async_load_b128(row0 + tid * 4, &rowbuf[0][0] + tid * 4);
  if (Tb > 1) {
    const float* row1 = log_probs + ((size_t)1 * B + b) * C;
    async_load_b128(row1 + tid * 4, &rowbuf[1][0] + tid * 4);
  }

  // --- exclusive cumsum offset for this batch's ragged targets
  if (tid == 0) {
    int off = 0;
    for (int i = 0; i < b; ++i) off += target_lengths[i];
    offset_sh = off;
  }
  __syncthreads();
  const int off = offset_sh;

  // --- extended targets (blank-interleaved) + skip condition
  for (int s = tid; s < S; s += TPB) {
    int e = BLANK_;
    if (s & 1) { int l = s >> 1; e = (l < Lb) ? targets[off + l] : BLANK_; }
    ext[s]   = e;
    alpha[s] = NEG_INF_F;
  }
  __syncthreads();
  for (int s = tid; s < S; s += TPB) {
    int e  = ext[s];
    int e2 = (s >= 2) ? ext[s - 2] : -1;
    skipc[s] = (e != BLANK_ && e != e2) ? 1 : 0;
  }

  // --- t = 0 initialization from row 0
  wait_async_le((Tb > 1) ? 1 : 0);   // row 0 landed in LDS
  __syncthreads();
  if (tid < 2) alpha[tid] = rowbuf[0][ext[tid]];
  __syncthreads();                   // row-0 reads done before overwrite
  if (2 <= Tb - 1) {
    const float* row2 = log_probs + ((size_t)2 * B + b) * C;
    async_load_b128(row2 + tid * 4, &rowbuf[0][0] + tid * 4);
  }

  // --- main recurrence, t = 1 .. Tb-1
  const bool active = (tid < S);
  for (int t = 1; t < Tb; ++t) {
    wait_async_le((t < Tb - 1) ? 1 : 0);  // oldest in-flight row (== t) done
    __syncthreads();                      // publish row t + previous alpha
    float newv = NEG_INF_F;
    if (active) {
      float a   = alpha[tid];
      float am1 = (tid >= 1) ? alpha[tid - 1] : NEG_INF_F;
      float lp  = lae(a, am1);
      if (skipc[tid]) {
        float am2 = (tid >= 2) ? alpha[tid - 2] : NEG_INF_F;
        lp = lae(lp, am2);
      }
      float em = rowbuf[t & 1][ext[tid]];
      newv = lp + em;
    }
    __syncthreads();                      // all alpha/rowbuf reads retired
    if (active) alpha[tid] = newv;
    if (t + 2 <= Tb - 1) {                // stream row t+2 into freed buffer
      const float* rw = log_probs + ((size_t)(t + 2) * B + b) * C;
      async_load_b128(rw + tid * 4, &rowbuf[t & 1][0] + tid * 4);
    }
  }
  __syncthreads();

  // --- per-batch NLL
  if (tid == 0) {
    float last   = alpha[2 * Lb];
    float second = alpha[(2 * Lb - 1 > 0) ? (2 * Lb - 1) : 0];
    float total  = (Lb > 0) ? lae(last, second) : last;
    float nll    = -total;
    if (Lb > 0) nll /= (float)Lb;
    nll_out[b] = nll;
  }
}

// ---------------------------------------------------------------------------
// Mean over B via V_WMMA_F32_16X16X4_F32 with A == ones:
// D[m][n] = sum_k B[k][n]  =>  sum over one D row (16 lanes) = exact fp32 sum
// of all 64 values fed into the B operand, independent of lane packing.
// Single wave, fixed order => deterministic.
// ---------------------------------------------------------------------------
__global__ __launch_bounds__(32)
void CTC_reduce_kernel(const float* __restrict__ nll, float* __restrict__ out, int B) {
  const int lane = threadIdx.x;  // 0..31
#if __has_builtin(__builtin_amdgcn_wmma_f32_16x16x4_f32)
  v8f acc = {};
  v2f aones; aones.x = 1.0f; aones.y = 1.0f;
  for (int base = 0; base < B; base += 64) {
    v2f bv;
    int i0 = base + lane;
    int i1 = base + 32 + lane;
    bv.x = (i0 < B) ? nll[i0] : 0.0f;
    bv.y = (i1 < B) ? nll[i1] : 0.0f;
    acc = __builtin_amdgcn_wmma_f32_16x16x4_f32(
        /*neg_a=*/false, aones, /*neg_b=*/false, bv,
        /*c_mod=*/(short)0, acc, /*reuse_a=*/false, /*reuse_b=*/false);
  }
  float v = acc[0];                       // D[row 0 or 8][n] per lane
  v += __shfl_xor(v, 8, 32);              // sum n = 0..15 within each half
  v += __shfl_xor(v, 4, 32);
  v += __shfl_xor(v, 2, 32);
  v += __shfl_xor(v, 1, 32);
  if (lane == 0) out[0] = v / (float)B;
#else
  float v = 0.0f;
  for (int i = lane; i < B; i += 32) v += nll[i];
  v += __shfl_xor(v, 16, 32);
  v += __shfl_xor(v, 8, 32);
  v += __shfl_xor(v, 4, 32);
  v += __shfl_xor(v, 2, 32);
  v += __shfl_xor(v, 1, 32);
  if (lane == 0) out[0] = v / (float)B;
#endif
}

// ---------------------------------------------------------------------------
extern "C" void kernel_launch(void* const* d_in, const int* in_sizes, int n_in,
                              void* d_out, int out_size, void* d_ws, size_t ws_size,
                              hipStream_t stream) {
  const float* log_probs      = (const float*)d_in[0];
  const int*   targets        = (const int*)d_in[1];
  const int*   input_lengths  = (const int*)d_in[2];
  const int*   target_lengths = (const int*)d_in[3];

  const int B    = in_sizes[2];
  const int Lmax = in_sizes[1] / B;      // 64
  const int C    = MAX_C;                // 1024 (reference-fixed)
  const int T    = (int)((long long)in_sizes[0] / ((long long)B * C)); // 512

  float* nll_ws = (float*)d_ws;          // B floats of scratch

  CTC_alpha_kernel<<<B, TPB, 0, stream>>>(log_probs, targets, input_lengths,
                                          target_lengths, nll_ws, T, B, C, Lmax);
  CTC_reduce_kernel<<<1, 32, 0, stream>>>(nll_ws, (float*)d_out, B);
}